// CorrelationImage_33406255629127
// MI455X (gfx1250) — compile-verified
//
#include <hip/hip_runtime.h>

// ---------------- problem dimensions (fixed by the reference) ----------------
#define BATCH        64
#define N_PER_BATCH  (64 * 64 * 64)          // C*H*W = 262144 floats per batch
#define F4_PER_BATCH (N_PER_BATCH / 4)       // 65536 float4 per batch
#define PB           16                      // reduction blocks per batch
#define SEG          (F4_PER_BATCH / PB)     // 4096 float4 per block
#define TPB          256                     // 8 waves (wave32)

typedef __attribute__((ext_vector_type(2))) float v2f;
typedef __attribute__((ext_vector_type(8))) float v8f;

// Full-precision wave32 reduction through the matrix pipe:
// A (16x4 f32): a0 = p (lane partial), a1 = 0
//   -> A[m,0] = p_m, A[m,2] = p_{m+16}, A[m,1]=A[m,3]=0
// B (4x16) = ones  =>  D[m,n] = p_m + p_{m+16}  (independent of n)
// Each lane holds 8 rows of D (C/D layout) -> summing its 8 VGPRs folds 16
// partials; one shfl_xor(16) finishes the 32-lane sum. fp32 RNE throughout.
__device__ __forceinline__ float wave_reduce_wmma(float p) {
    v2f a; a[0] = p;   a[1] = 0.0f;
    v2f b; b[0] = 1.0f; b[1] = 1.0f;
    v8f c = {};
    c = __builtin_amdgcn_wmma_f32_16x16x4_f32(false, a, false, b,
                                              (short)0, c, false, false);
    float s = ((c[0] + c[1]) + (c[2] + c[3])) + ((c[4] + c[5]) + (c[6] + c[7]));
    s += __shfl_xor(s, 16, 32);   // combine the two half-wave row groups
    return s;
}

// Pass 1: per-batch partial dot products. grid = (PB, BATCH), block = 256.
// Streams 2 x 64 KiB per block with b128 loads; VALU FMA is nowhere near the
// bottleneck (HBM-bound), WMMA handles the cross-lane fold for free.
__global__ void __launch_bounds__(TPB)
corr_dot_kernel(const float4* __restrict__ m1, const float4* __restrict__ m2,
                float* __restrict__ partials) {
    const int blk   = blockIdx.x;   // 0..PB-1
    const int batch = blockIdx.y;   // 0..BATCH-1
    const int t     = threadIdx.x;
    const long base = (long)batch * F4_PER_BATCH + (long)blk * SEG;

    float p = 0.0f;
#pragma unroll
    for (int k = 0; k < SEG / TPB; ++k) {
        const float4 x = m1[base + k * TPB + t];
        const float4 y = m2[base + k * TPB + t];
        p = fmaf(x.x, y.x, p);
        p = fmaf(x.y, y.y, p);
        p = fmaf(x.z, y.z, p);
        p = fmaf(x.w, y.w, p);
    }

    const float wsum = wave_reduce_wmma(p);   // EXEC all-ones here (no divergence)

    __shared__ float red[TPB / 32];
    const int wave = t >> 5;
    if ((t & 31) == 0) red[wave] = wsum;
    __syncthreads();
    if (t == 0) {
        float s = 0.0f;
#pragma unroll
        for (int w = 0; w < TPB / 32; ++w) s += red[w];
        partials[batch * PB + blk] = s;
    }
}

// Pass 2: tiny. corr[b] = sum of its PB partials; factor[b] = 1 - corr[b]/||corr||.
__global__ void __launch_bounds__(BATCH)
corr_norm_kernel(const float* __restrict__ partials, float* __restrict__ factor) {
    __shared__ float corr_s[BATCH];
    __shared__ float rnorm_s;
    const int b = threadIdx.x;

    float c = 0.0f;
#pragma unroll
    for (int i = 0; i < PB; ++i) c += partials[b * PB + i];
    corr_s[b] = c;
    __syncthreads();

    if (b == 0) {
        float ss = 0.0f;
#pragma unroll
        for (int i = 0; i < BATCH; ++i) ss += corr_s[i] * corr_s[i];
        rnorm_s = 1.0f / sqrtf(ss);
    }
    __syncthreads();

    factor[b] = 1.0f - corr_s[b] * rnorm_s;
}

// Pass 3: out = map1 + map2 * factor[batch]. Inputs are L2-resident (128 MiB
// fits the 192 MB L2), so this pass runs near L2 read + HBM write speed.
__global__ void __launch_bounds__(TPB)
corr_apply_kernel(const float4* __restrict__ m1, const float4* __restrict__ m2,
                  const float* __restrict__ factor, float4* __restrict__ out) {
    const int i = blockIdx.x * TPB + threadIdx.x;       // float4 index
    const float f = factor[i >> 16];                    // i / F4_PER_BATCH
    const float4 x = m1[i];
    const float4 y = m2[i];
    float4 o;
    o.x = fmaf(y.x, f, x.x);
    o.y = fmaf(y.y, f, x.y);
    o.z = fmaf(y.z, f, x.z);
    o.w = fmaf(y.w, f, x.w);
    out[i] = o;
}

extern "C" void kernel_launch(void* const* d_in, const int* in_sizes, int n_in,
                              void* d_out, int out_size, void* d_ws, size_t ws_size,
                              hipStream_t stream) {
    (void)in_sizes; (void)n_in; (void)out_size; (void)ws_size;
    const float4* m1 = (const float4*)d_in[0];
    const float4* m2 = (const float4*)d_in[1];

    float* partials = (float*)d_ws;            // BATCH * PB floats
    float* factor   = partials + BATCH * PB;   // BATCH floats

    corr_dot_kernel<<<dim3(PB, BATCH), TPB, 0, stream>>>(m1, m2, partials);
    corr_norm_kernel<<<1, BATCH, 0, stream>>>(partials, factor);

    const int total_f4 = BATCH * F4_PER_BATCH;           // 4,194,304
    corr_apply_kernel<<<total_f4 / TPB, TPB, 0, stream>>>(m1, m2, factor,
                                                          (float4*)d_out);
}